// MPPI_76519137345985
// MI455X (gfx1250) — compile-verified
//
#include <hip/hip_runtime.h>
#include <math.h>
#include <stdint.h>

// ---------------- problem constants (match reference) ----------------
#define K_ROLL   65536
#define T_STEPS  96
#define NU       2
#define BLK      256        // threads per block (8 waves of 32)
#define TC       16         // timesteps staged per chunk
#define NCH      (T_STEPS / TC)   // 6 chunks
#define BEV_PX   50
#define BEV_N    (BEV_PX * BEV_PX)

#define F_MAX_CURV_RATE  0.48108237397409f   // tan(30/57.3)/0.48 * 20 * 0.02
#define F_WSPD           17.0f
#define F_MAX_SPEED_RATE 0.196f              // 9.8 * 0.02
#define F_DT             0.02f
#define F_BEV_RES        0.4f

#define AS1 __attribute__((address_space(1)))
#define AS3 __attribute__((address_space(3)))

typedef int v2i __attribute__((ext_vector_type(2)));

// CDNA5 async global->LDS path (gfx1250). Guarded so compile never fails.
#if defined(__has_builtin)
#if __has_builtin(__builtin_amdgcn_global_load_async_to_lds_b64) && \
    __has_builtin(__builtin_amdgcn_s_wait_asynccnt)
#define HAVE_ASYNC 1
#endif
#endif
#ifndef HAVE_ASYNC
#define HAVE_ASYNC 0
#endif

__device__ __forceinline__ float clampf(float v, float lim) {
    return fminf(fmaxf(v, -lim), lim);
}

// Stage one T-chunk of this block's 256 noise rows into an LDS tile.
// Tile layout: buf[(tsub*BLK + k_local)*2 + u]  -> conflict-free float2 reads.
// Global: each row's chunk is 128 contiguous bytes; lanes 0..15 of a wave copy
// one row (16 x b64), so every wave issues fully-coalesced requests.
__device__ __forceinline__ void issue_chunk(const float* __restrict__ nbase,
                                            int ch, float* buf, int tid) {
#pragma unroll
    for (int j = 0; j < TC; ++j) {
        int L  = tid + BLK * j;       // 0..4095
        int kl = L >> 4;              // local rollout row 0..255
        int i  = L & 15;              // timestep-within-chunk 0..15
        const float* g = nbase + (size_t)kl * (T_STEPS * NU) + (size_t)ch * (TC * NU) + i * NU;
        float* d = buf + ((size_t)i * BLK + kl) * NU;
#if HAVE_ASYNC
        __builtin_amdgcn_global_load_async_to_lds_b64(
            (AS1 v2i*)(v2i*)g, (AS3 v2i*)(v2i*)d, 0, 0);
#else
        d[0] = g[0];
        d[1] = g[1];
#endif
    }
}

// ---------------- K1: rollout + per-sample cost ----------------
__global__ __launch_bounds__(BLK) void mppi_rollout(
    const float* __restrict__ state, const float* __restrict__ U,
    const float* __restrict__ noise, const float* __restrict__ sigInv,
    const float* __restrict__ bev,   const float* __restrict__ goal,
    float* __restrict__ cost_out)
{
    __shared__ float sU[T_STEPS * NU];          // rolled U
    __shared__ float sW[T_STEPS * NU];          // W[t][v] = sum_u sigInv[v][u]*Ur[t][u]
    __shared__ float sMap[BEV_N];               // 10 KB BEV map
    __shared__ float tile[2][TC * BLK * NU];    // double-buffered noise tiles (2x32KB)

    const int tid = threadIdx.x;
    const int k   = blockIdx.x * BLK + tid;

    for (int i = tid; i < BEV_N; i += BLK) sMap[i] = bev[i];
    if (tid < T_STEPS) {
        float u0 = (tid < T_STEPS - 1) ? U[(tid + 1) * NU + 0] : 0.0f;
        float u1 = (tid < T_STEPS - 1) ? U[(tid + 1) * NU + 1] : 0.0f;
        sU[tid * NU + 0] = u0;
        sU[tid * NU + 1] = u1;
        sW[tid * NU + 0] = sigInv[0] * u0 + sigInv[1] * u1;
        sW[tid * NU + 1] = sigInv[2] * u0 + sigInv[3] * u1;
    }

    const float x0 = state[0], y0 = state[1], yaw0 = state[5];
    const float gx = goal[0],  gy = goal[1];
    const float* nbase = noise + (size_t)blockIdx.x * BLK * T_STEPS * NU;

    issue_chunk(nbase, 0, tile[0], tid);

    float cacc = 0.0f, sacc = 0.0f;
    float yaw = yaw0, xx = x0, yy = y0, cost = 0.0f;
    float pc = 0.0f, ps = 0.0f, pn0 = 0.0f, pn1 = 0.0f;  // lookbehind state

    auto step = [&](float cu, float su, float n0, float n1, int tt, bool term) {
        yaw += su * cu * F_DT;
        float sn, cs;
        sincosf(yaw, &sn, &cs);
        xx += su * cs * F_DT;
        yy += su * sn * F_DT;
        int ix = (int)(xx / F_BEV_RES + (float)(BEV_PX / 2));  // trunc == astype(int32)
        int iy = (int)(yy / F_BEV_RES + (float)(BEV_PX / 2));
        ix = ix < 0 ? 0 : (ix > BEV_PX - 1 ? BEV_PX - 1 : ix);
        iy = iy < 0 ? 0 : (iy > BEV_PX - 1 ? BEV_PX - 1 : iy);
        float dxv = xx - gx, dyv = yy - gy;
        float dist = sqrtf(dxv * dxv + dyv * dyv);
        float c = sMap[iy * BEV_PX + ix] + 0.1f * dist
                + n0 * sW[tt * NU + 0] + n1 * sW[tt * NU + 1];
        if (term) c += dist;   // terminal cost
        cost += c;
    };

    for (int ch = 0; ch < NCH; ++ch) {
        const float* buf = tile[ch & 1];
        if (ch + 1 < NCH) issue_chunk(nbase, ch + 1, tile[(ch + 1) & 1], tid);
#if HAVE_ASYNC
        if (ch + 1 < NCH) __builtin_amdgcn_s_wait_asynccnt(TC); // current chunk done (in-order)
        else              __builtin_amdgcn_s_wait_asynccnt(0);
#endif
        __syncthreads();   // staged data (and map/U on first iter) visible to all waves
#pragma unroll
        for (int ts = 0; ts < TC; ++ts) {
            int t = ch * TC + ts;
            float n0 = buf[((size_t)ts * BLK + tid) * NU + 0];
            float n1 = buf[((size_t)ts * BLK + tid) * NU + 1];
            float rc = (sU[t * NU + 0] + n0) * F_MAX_CURV_RATE;
            float rs = (sU[t * NU + 1] + n1) * F_WSPD;
            if (t > 0) {
                cacc += clampf(rc - pc, F_MAX_CURV_RATE);
                sacc += clampf(rs - ps, F_MAX_SPEED_RATE);
                step(cacc, sacc, pn0, pn1, t - 1, false);
            }
            pc = rc; ps = rs; pn0 = n0; pn1 = n1;
        }
        __syncthreads();   // everyone done reading buf before it is refilled
    }
    // last timestep uses the RAW (unlimited) values, per reference concat
    step(pc, ps, pn0, pn1, T_STEPS - 1, true);

    cost_out[k] = cost;
}

// ---------------- K2: global min (beta), single block, deterministic ------
__global__ __launch_bounds__(1024) void mppi_min(const float* __restrict__ cost,
                                                 float* __restrict__ beta) {
    __shared__ float red[1024];
    float m = 3.402823e38f;
    for (int i = threadIdx.x; i < K_ROLL; i += 1024) m = fminf(m, cost[i]);
    red[threadIdx.x] = m;
    __syncthreads();
    for (int s = 512; s > 0; s >>= 1) {
        if (threadIdx.x < s) red[threadIdx.x] = fminf(red[threadIdx.x], red[threadIdx.x + s]);
        __syncthreads();
    }
    if (threadIdx.x == 0) *beta = red[0];
}

// ---------------- K3: softmax weights + per-block partial sums -----------
__global__ __launch_bounds__(256) void mppi_weight(const float* __restrict__ cost,
                                                   const float* __restrict__ beta,
                                                   const float* __restrict__ noise,
                                                   float* __restrict__ partial) {
    __shared__ float r0[256], r1[256], r2[256];
    const int tid = threadIdx.x;
    const int k   = blockIdx.x * 256 + tid;
    const float b = *beta;
    float w  = expf(-(cost[k] - b));                   // LAMBDA == 1
    float n0 = noise[(size_t)k * T_STEPS * NU + 0];    // t = 0 row only
    float n1 = noise[(size_t)k * T_STEPS * NU + 1];
    r0[tid] = w; r1[tid] = w * n0; r2[tid] = w * n1;
    __syncthreads();
    for (int s = 128; s > 0; s >>= 1) {
        if (tid < s) {
            r0[tid] += r0[tid + s];
            r1[tid] += r1[tid + s];
            r2[tid] += r2[tid + s];
        }
        __syncthreads();
    }
    if (tid == 0) {
        partial[blockIdx.x * 3 + 0] = r0[0];
        partial[blockIdx.x * 3 + 1] = r1[0];
        partial[blockIdx.x * 3 + 2] = r2[0];
    }
}

// ---------------- K4: final reduce + output ------------------------------
__global__ __launch_bounds__(256) void mppi_final(const float* __restrict__ partial,
                                                  const float* __restrict__ U,
                                                  float* __restrict__ out) {
    __shared__ float r0[256], r1[256], r2[256];
    const int tid = threadIdx.x;
    r0[tid] = partial[tid * 3 + 0];
    r1[tid] = partial[tid * 3 + 1];
    r2[tid] = partial[tid * 3 + 2];
    __syncthreads();
    for (int s = 128; s > 0; s >>= 1) {
        if (tid < s) {
            r0[tid] += r0[tid + s];
            r1[tid] += r1[tid + s];
            r2[tid] += r2[tid + s];
        }
        __syncthreads();
    }
    if (tid == 0) {
        float sw = r0[0];
        // rolled-U row 0 == original U[1]
        out[0] = U[1 * NU + 0] + r1[0] / sw;
        out[1] = U[1 * NU + 1] + r2[0] / sw;
    }
}

extern "C" void kernel_launch(void* const* d_in, const int* in_sizes, int n_in,
                              void* d_out, int out_size, void* d_ws, size_t ws_size,
                              hipStream_t stream) {
    (void)in_sizes; (void)n_in; (void)out_size; (void)ws_size;
    const float* state  = (const float*)d_in[0];
    const float* U      = (const float*)d_in[1];
    const float* noise  = (const float*)d_in[2];
    const float* sigInv = (const float*)d_in[3];
    const float* bev    = (const float*)d_in[4];
    const float* goal   = (const float*)d_in[5];
    // d_in[6] (last_action) provably does not affect the output.

    float* ws      = (float*)d_ws;
    float* cost    = ws;                    // K floats
    float* beta    = ws + K_ROLL;           // 1 float
    float* partial = ws + K_ROLL + 8;       // 256*3 floats

    mppi_rollout<<<K_ROLL / BLK, BLK, 0, stream>>>(state, U, noise, sigInv, bev, goal, cost);
    mppi_min   <<<1, 1024, 0, stream>>>(cost, beta);
    mppi_weight<<<K_ROLL / 256, 256, 0, stream>>>(cost, beta, noise, partial);
    mppi_final <<<1, 256, 0, stream>>>(partial, U, (float*)d_out);
}